// ForwardKinematics_51694226374939
// MI455X (gfx1250) — compile-verified
//
#include <hip/hip_runtime.h>

// Forward kinematics: B=64, J=24, T=4096
//   r6d   [B, J, 6, T] f32
//   trs   [B, 3, T]    f32
//   offset[1, J, 3, 1] f32  (72 floats)
//   out   [B, J, 3, T] f32
//
// One lane per (b, t). T innermost => stride-1 coalesced vmem across the
// wave. Joint chain fully unrolled with a constexpr parent table so SROA
// keeps only ~3 live parent matrices in VGPRs. ~230 MB traffic => ~10 us
// at 23.3 TB/s (memory-bound); normalizations use v_rsq_f32 via
// 1/max(sqrt(d),1e-8) == rsqrt(max(d,1e-16)) to collapse the IEEE
// sqrt+fdiv expansions (~35% VALU reduction).
//
// CDNA5-specific: offset table staged to LDS via async global->LDS DMA
// (global_load_async_to_lds_b32 + s_wait_asynccnt split counter).

#define NJ 24
#define NT 4096
#define NB 64

__device__ __forceinline__ float fast_rnorm(float d) {
    // == 1.0f / fmaxf(sqrtf(d), 1e-8f) for d >= 0
#if defined(__gfx1250__) && __has_builtin(__builtin_amdgcn_rsqf)
    return __builtin_amdgcn_rsqf(fmaxf(d, 1e-16f));
#else
    return 1.0f / fmaxf(sqrtf(d), 1e-8f);
#endif
}

__global__ __launch_bounds__(256) void
fk_kernel(const float* __restrict__ r6d,
          const float* __restrict__ trs,
          const float* __restrict__ offset,
          float* __restrict__ out)
{
    __shared__ float soff[NJ * 3];
    const int tid = threadIdx.x;

    // ---- stage 72-float offset table into LDS (async DMA on gfx1250) ----
#if defined(__gfx1250__) && __has_builtin(__builtin_amdgcn_global_load_async_to_lds_b32)
    if (tid < NJ * 3) {
        typedef __attribute__((address_space(1))) int gint;
        typedef __attribute__((address_space(3))) int lint;
        __builtin_amdgcn_global_load_async_to_lds_b32(
            (gint*)(offset + tid),
            (lint*)(&soff[tid]),
            /*offset=*/0, /*cpol=*/0);
#if __has_builtin(__builtin_amdgcn_s_wait_asynccnt)
        __builtin_amdgcn_s_wait_asynccnt(0);
#else
        asm volatile("s_wait_asynccnt 0" ::: "memory");
#endif
    }
#else
    if (tid < NJ * 3) soff[tid] = offset[tid];
#endif
    __syncthreads();

    const int g = blockIdx.x * 256 + tid;   // 0 .. NB*NT-1
    const int b = g >> 12;                  // g / NT
    const int t = g & (NT - 1);             // g % NT

    const float* r6 = r6d + (size_t)b * NJ * 6 * NT + t;
    const float* tr = trs + (size_t)b * 3 * NT + t;
    float*       ob = out + (size_t)b * NJ * 3 * NT + t;

    const float rt0 = tr[0 * NT];
    const float rt1 = tr[1 * NT];
    const float rt2 = tr[2 * NT];

    constexpr int P[NJ] = {-1, 0, 0, 0, 1, 2, 3, 4, 5, 6, 7, 8,
                            9, 9, 9, 12, 13, 14, 16, 17, 18, 19, 20, 21};

    // Fully-unrolled chain: constant indices -> SROA keeps these in VGPRs
    // with liveness pruning (max ~3 parent matrices live at once).
    float gm[NJ][9];
    float ps[NJ][3];

#pragma unroll
    for (int j = 0; j < NJ; ++j) {
        const float* rj = r6 + (size_t)j * 6 * NT;
        const float a10 = rj[0 * NT], a11 = rj[1 * NT], a12 = rj[2 * NT];
        const float a20 = rj[3 * NT], a21 = rj[4 * NT], a22 = rj[5 * NT];

        // b1 = a1 / max(|a1|, 1e-8)  ==  a1 * rsqrt(max(a1.a1, 1e-16))
        const float i1  = fast_rnorm(a10 * a10 + a11 * a11 + a12 * a12);
        const float b10 = a10 * i1, b11 = a11 * i1, b12 = a12 * i1;

        // a2p = a2 - (b1.a2) b1 ; b2 = a2p / max(|a2p|, 1e-8)
        const float d   = b10 * a20 + b11 * a21 + b12 * a22;
        const float p0  = a20 - d * b10;
        const float p1  = a21 - d * b11;
        const float p2  = a22 - d * b12;
        const float i2  = fast_rnorm(p0 * p0 + p1 * p1 + p2 * p2);
        const float b20 = p0 * i2, b21 = p1 * i2, b22 = p2 * i2;

        // b3 = b1 x b2
        const float b30 = b11 * b22 - b12 * b21;
        const float b31 = b12 * b20 - b10 * b22;
        const float b32 = b10 * b21 - b11 * b20;

        // local rotation, row-major rows (b1, b2, b3)
        const float m[9] = {b10, b11, b12, b20, b21, b22, b30, b31, b32};

        const int p = P[j];
        if (p < 0) {
#pragma unroll
            for (int k = 0; k < 9; ++k) gm[j][k] = m[k];
            ps[j][0] = rt0; ps[j][1] = rt1; ps[j][2] = rt2;
        } else {
            // gm[j] = gm[p] * m   (3x3 matmul)
#pragma unroll
            for (int r = 0; r < 3; ++r) {
#pragma unroll
                for (int c = 0; c < 3; ++c) {
                    gm[j][r * 3 + c] = gm[p][r * 3 + 0] * m[0 * 3 + c]
                                     + gm[p][r * 3 + 1] * m[1 * 3 + c]
                                     + gm[p][r * 3 + 2] * m[2 * 3 + c];
                }
            }
            // ps[j] = gm[p] * off[j] + ps[p]
            const float o0 = soff[j * 3 + 0];
            const float o1 = soff[j * 3 + 1];
            const float o2 = soff[j * 3 + 2];
#pragma unroll
            for (int r = 0; r < 3; ++r) {
                ps[j][r] = gm[p][r * 3 + 0] * o0
                         + gm[p][r * 3 + 1] * o1
                         + gm[p][r * 3 + 2] * o2 + ps[p][r];
            }
        }

        float* oj = ob + (size_t)j * 3 * NT;
        oj[0 * NT] = ps[j][0];
        oj[1 * NT] = ps[j][1];
        oj[2 * NT] = ps[j][2];
    }
}

extern "C" void kernel_launch(void* const* d_in, const int* in_sizes, int n_in,
                              void* d_out, int out_size, void* d_ws, size_t ws_size,
                              hipStream_t stream) {
    const float* r6d = (const float*)d_in[0];
    const float* trs = (const float*)d_in[1];
    const float* off = (const float*)d_in[2];
    float* out = (float*)d_out;

    const int total = NB * NT;            // 262,144 lanes
    dim3 grid(total / 256), block(256);   // 1024 blocks x 8 waves
    hipLaunchKernelGGL(fk_kernel, grid, block, 0, stream, r6d, trs, off, out);
}